// GlobalPointer_11201274708616
// MI455X (gfx1250) — compile-verified
//
#include <hip/hip_runtime.h>

typedef __attribute__((ext_vector_type(2))) float v2f;
typedef __attribute__((ext_vector_type(8))) float v8f;

#define NB   8
#define NS   512
#define NH   768
#define NL   16
#define ND   64
#define BIGNEG 1000000000000.0f

// D = A(16x4) * B(4x16) + C, fp32 WMMA
__device__ __forceinline__ v8f wmma_f32(v2f a, v2f b, v8f c) {
  return __builtin_amdgcn_wmma_f32_16x16x4_f32(false, a, false, b, (short)0, c,
                                               false, false);
}

// ---------------------------------------------------------------------------
// Kernel 1: out = seq @ W^T + bias, fused RoPE, scatter into q/k (B,L,S,D)
// M = B*S = 4096, N = L*2D = 2048, K = H = 768
// Block: 256 thr (8 waves), tile 128(M) x 64(N); wave = 32x32 (2x2 WMMA)
// ---------------------------------------------------------------------------
__global__ __launch_bounds__(256) void proj_rope_kernel(
    const float* __restrict__ seq, const float* __restrict__ Wmat,
    const float* __restrict__ bias, float* __restrict__ qbuf,
    float* __restrict__ kbuf) {
  __shared__ float As[128][20];  // K-tile 16, padded
  __shared__ float Bs[64][20];

  const int tid  = threadIdx.x;
  const int lane = tid & 31;
  const int wave = tid >> 5;
  const int wm   = wave >> 1;   // 0..3
  const int wn   = wave & 1;    // 0..1
  const int grp  = lane >> 4;   // 0..1
  const int lrow = lane & 15;

  const int m0 = blockIdx.y * 128;
  const int n0 = blockIdx.x * 64;

  v8f acc[2][2] = {};

  for (int kt = 0; kt < NH; kt += 16) {
    // stage A: 128 rows x 16 k (2 x float4 per thread)
#pragma unroll
    for (int j = 0; j < 2; ++j) {
      int c   = tid + 256 * j;
      int row = c >> 2;
      int kq  = (c & 3) * 4;
      float4 v = *(const float4*)(seq + (size_t)(m0 + row) * NH + kt + kq);
      *(float4*)&As[row][kq] = v;
    }
    // stage B (= W rows): 64 x 16 (1 x float4 per thread)
    {
      int row = tid >> 2;
      int kq  = (tid & 3) * 4;
      float4 v = *(const float4*)(Wmat + (size_t)(n0 + row) * NH + kt + kq);
      *(float4*)&Bs[row][kq] = v;
    }
    __syncthreads();

#pragma unroll
    for (int kc = 0; kc < 16; kc += 4) {
      const int c0 = kc + 2 * grp;
      v2f a0, a1, b0, b1;
      a0.x = As[wm * 32 + lrow][c0];      a0.y = As[wm * 32 + lrow][c0 + 1];
      a1.x = As[wm * 32 + 16 + lrow][c0]; a1.y = As[wm * 32 + 16 + lrow][c0 + 1];
      b0.x = Bs[wn * 32 + lrow][c0];      b0.y = Bs[wn * 32 + lrow][c0 + 1];
      b1.x = Bs[wn * 32 + 16 + lrow][c0]; b1.y = Bs[wn * 32 + 16 + lrow][c0 + 1];
      acc[0][0] = wmma_f32(a0, b0, acc[0][0]);
      acc[0][1] = wmma_f32(a0, b1, acc[0][1]);
      acc[1][0] = wmma_f32(a1, b0, acc[1][0]);
      acc[1][1] = wmma_f32(a1, b1, acc[1][1]);
    }
    __syncthreads();
  }

  // Epilogue: bias + interleaved RoPE, write q/k in (B,L,S,D) layout.
#pragma unroll
  for (int ti = 0; ti < 2; ++ti) {
#pragma unroll
    for (int tj = 0; tj < 2; ++tj) {
      const int mBase = m0 + wm * 32 + ti * 16;
      const int n     = n0 + wn * 32 + tj * 16 + lrow;  // column in [0,2048)
      const int l     = n >> 7;
      const int rr    = n & 127;
      const int half  = rr >> 6;   // 0 = q, 1 = k
      const int d     = rr & 63;
      // theta_i = 1000^(-2i/64), 2i = d&~1
      const float theta =
          __expf(-(float)(d & ~1) * (6.9077552789821368f / 64.0f));
      float* dst = half ? kbuf : qbuf;
      const float bn = bias[n];
#pragma unroll
      for (int r = 0; r < 8; ++r) {
        const int m  = mBase + r + 8 * grp;   // global row in [0,4096)
        const int bb = m >> 9;
        const int s  = m & (NS - 1);
        float val = acc[ti][tj][r] + bn;
        float sn, cs;
        __sincosf((float)s * theta, &sn, &cs);
        // rotate-half partner lives in the adjacent lane (adjacent column)
        float p = __shfl_xor(val, 1, 32);
        float o = (d & 1) ? fmaf(val, cs, p * sn) : fmaf(val, cs, -p * sn);
        dst[(((size_t)(bb * NL + l)) * NS + s) * ND + d] = o;
      }
    }
  }
}

// ---------------------------------------------------------------------------
// Kernel 2: per (b,l): logits = q(512x64) @ k(512x64)^T, mask + tril + scale
// Block: 256 thr, tile 128(M) x 64(N), K = 64 staged once in LDS.
// ---------------------------------------------------------------------------
__global__ __launch_bounds__(256) void gp_logits_kernel(
    const float* __restrict__ qbuf, const float* __restrict__ kbuf,
    const int* __restrict__ mask, float* __restrict__ out) {
  __shared__ float Qs[128][68];
  __shared__ float Ks[64][68];

  const int tid  = threadIdx.x;
  const int lane = tid & 31;
  const int wave = tid >> 5;
  const int wm   = wave >> 1;
  const int wn   = wave & 1;
  const int grp  = lane >> 4;
  const int lrow = lane & 15;

  const int bl = blockIdx.z;       // b*L + l
  const int bb = bl >> 4;
  const int m0 = blockIdx.y * 128;
  const int n0 = blockIdx.x * 64;

  const float* qbase = qbuf + (size_t)bl * NS * ND;
  const float* kbase = kbuf + (size_t)bl * NS * ND;

  // stage Q 128x64 (8 x float4 per thread) and K 64x64 (4 x float4)
#pragma unroll
  for (int j = 0; j < 8; ++j) {
    int c   = tid + 256 * j;
    int row = c >> 4;
    int kq  = (c & 15) * 4;
    float4 v = *(const float4*)(qbase + (size_t)(m0 + row) * ND + kq);
    *(float4*)&Qs[row][kq] = v;
  }
#pragma unroll
  for (int j = 0; j < 4; ++j) {
    int c   = tid + 256 * j;
    int row = c >> 4;
    int kq  = (c & 15) * 4;
    float4 v = *(const float4*)(kbase + (size_t)(n0 + row) * ND + kq);
    *(float4*)&Ks[row][kq] = v;
  }
  __syncthreads();

  v8f acc[2][2] = {};
#pragma unroll
  for (int kc = 0; kc < ND; kc += 4) {
    const int c0 = kc + 2 * grp;
    v2f a0, a1, b0, b1;
    a0.x = Qs[wm * 32 + lrow][c0];      a0.y = Qs[wm * 32 + lrow][c0 + 1];
    a1.x = Qs[wm * 32 + 16 + lrow][c0]; a1.y = Qs[wm * 32 + 16 + lrow][c0 + 1];
    b0.x = Ks[wn * 32 + lrow][c0];      b0.y = Ks[wn * 32 + lrow][c0 + 1];
    b1.x = Ks[wn * 32 + 16 + lrow][c0]; b1.y = Ks[wn * 32 + 16 + lrow][c0 + 1];
    acc[0][0] = wmma_f32(a0, b0, acc[0][0]);
    acc[0][1] = wmma_f32(a0, b1, acc[0][1]);
    acc[1][0] = wmma_f32(a1, b0, acc[1][0]);
    acc[1][1] = wmma_f32(a1, b1, acc[1][1]);
  }

  // Epilogue: pad mask (over key axis n), strict-lower tril, /sqrt(D)
#pragma unroll
  for (int ti = 0; ti < 2; ++ti) {
#pragma unroll
    for (int tj = 0; tj < 2; ++tj) {
      const int mBase = m0 + wm * 32 + ti * 16;
      const int n     = n0 + wn * 32 + tj * 16 + lrow;
      const float pad = (float)mask[bb * NS + n];
#pragma unroll
      for (int r = 0; r < 8; ++r) {
        const int m = mBase + r + 8 * grp;
        float v = acc[ti][tj][r];
        v = v * pad - (1.0f - pad) * BIGNEG;
        if (n < m) v -= BIGNEG;
        out[((size_t)bl * NS + m) * NS + n] = v * 0.125f;
      }
    }
  }
}

extern "C" void kernel_launch(void* const* d_in, const int* in_sizes, int n_in,
                              void* d_out, int out_size, void* d_ws,
                              size_t ws_size, hipStream_t stream) {
  const float* seq  = (const float*)d_in[0];
  const int*   mask = (const int*)d_in[1];
  const float* Wmat = (const float*)d_in[2];
  const float* bias = (const float*)d_in[3];
  float* out = (float*)d_out;

  float* qbuf = (float*)d_ws;                                   // 16 MB
  float* kbuf = qbuf + (size_t)NB * NL * NS * ND;               // 16 MB

  dim3 g1((NL * 2 * ND) / 64, (NB * NS) / 128);                 // 32 x 32
  proj_rope_kernel<<<g1, 256, 0, stream>>>(seq, Wmat, bias, qbuf, kbuf);

  dim3 g2(NS / 64, NS / 128, NB * NL);                          // 8 x 4 x 128
  gp_logits_kernel<<<g2, 256, 0, stream>>>(qbuf, kbuf, mask, out);
}